// MRConvLayer_47880295416393
// MI455X (gfx1250) — compile-verified
//
#include <hip/hip_runtime.h>

// MRConv layer for gfx1250 (MI455X).
// Inputs (setup_inputs order):
//   d_in[0] = x             [16384, 128]  f32
//   d_in[1] = rel_pos_table [289, 128]    f32
//   d_in[2] = W             [256, 256]    f32
//   d_in[3] = b             [256]         f32
// Output: [16384, 256] f32.
//
// Workspace layout (floats): x2 (N*C) | sq (N) | nbr (N*K ints) | y (N*2C)
// total ~24.6 MB.

#define N_PTS 16384
#define C_IN  128
#define KNN   9
#define C_OUT 256
#define PANEL 512
#define TILES_PER_PANEL (PANEL / 16)   // 32 tiles, 8 per wave
#define KSTEPS (C_IN / 4)              // 32 wmma steps per 16x16 tile

typedef __attribute__((ext_vector_type(2))) float v2f;
typedef __attribute__((ext_vector_type(4))) float v4f;
typedef __attribute__((ext_vector_type(8))) float v8f;

// ---------------------------------------------------------------------------
// Kernel 1: squared norms of raw x + relative positional embedding add.
// One block of 128 threads per row.
// ---------------------------------------------------------------------------
__global__ void mrconv_prep_kernel(const float* __restrict__ x,
                                   const float* __restrict__ rpt,
                                   float* __restrict__ x2,
                                   float* __restrict__ sq) {
    const int n = blockIdx.x;
    const int c = threadIdx.x;
    const float v = x[n * C_IN + c];
    // rel index: n = i*C + j -> i - j + (C-1), range [0, 254]
    const int r = (n >> 7) - (n & 127) + (C_IN - 1);
    x2[n * C_IN + c] = v + rpt[r * C_IN + c];

    __shared__ float red[C_IN];
    red[c] = v * v;
    __syncthreads();
    #pragma unroll
    for (int s = C_IN / 2; s > 0; s >>= 1) {
        if (c < s) red[c] += red[c + s];
        __syncthreads();
    }
    if (c == 0) sq[n] = red[0];
}

// ---------------------------------------------------------------------------
// Kernel 2: brute-force kNN (K=9) on raw x using fp32 WMMA for x @ x^T.
// One block = 4 waves handles a 16-row strip. Per row j we rank by
// s[j] = sq[j] - 2*dot(i,j) (sq[i] is a per-row constant, irrelevant to rank).
// B-fragment loads are batched in groups of 16 so the load stream pipelines
// ahead of the WMMA chain instead of a full wait before every WMMA.
// ---------------------------------------------------------------------------
__launch_bounds__(128)
__global__ void mrconv_knn_kernel(const float* __restrict__ x,
                                  const float* __restrict__ sq,
                                  int* __restrict__ nbr) {
    __shared__ __align__(16) float Ds[16][PANEL + 4]; // dot tiles, bank-padded
    __shared__ float mv[16][8 * KNN];                 // merge buffers
    __shared__ int   mi[16][8 * KNN];

    const int rbase = blockIdx.x * 16;
    const int tid   = threadIdx.x;
    const int lane  = tid & 31;
    const int wave  = tid >> 5;

    // --- preload A fragments: 16 rows x 128 K, kept in VGPRs all kernel.
    // A layout (16x4 f32): lanes 0-15 hold K={0,1}, lanes 16-31 hold K={2,3}.
    const int arow = rbase + (lane & 15);
    const float* aptr = x + arow * C_IN + ((lane < 16) ? 0 : 2);
    v2f afrag[KSTEPS];
    #pragma unroll
    for (int k = 0; k < KSTEPS; ++k)
        afrag[k] = *(const v2f*)(aptr + 4 * k);

    // --- per-thread running top-9 (8 threads per row, thread scans 64 cols/panel)
    const int row  = tid >> 3;
    const int seg  = tid & 7;
    const int grow = rbase + row;
    float bv[KNN + 1];
    int   bi[KNN + 1];
    #pragma unroll
    for (int q = 0; q <= KNN; ++q) { bv[q] = 3.0e38f; bi[q] = 0; }

    for (int pb = 0; pb < N_PTS; pb += PANEL) {
        // ================= compute phase: WMMA dot tiles into LDS ==========
        #pragma unroll 1
        for (int t = wave; t < TILES_PER_PANEL; t += 4) {
            const int jb = pb + t * 16;
            // B layout (4x16 f32): VGPR0 = rows K0 / K2, VGPR1 = rows K1 / K3
            const float* bptr =
                x + (jb + (lane & 15)) * C_IN + ((lane < 16) ? 0 : 2);
            // warm cache one tile (this wave's next: +4 tiles = +64 rows) ahead
            __builtin_prefetch(bptr + 64 * C_IN, 0, 1);

            v8f acc = {};
            #pragma unroll
            for (int half = 0; half < 2; ++half) {
                v2f bfr[16];                 // batch loads -> pipelined waits
                #pragma unroll
                for (int k = 0; k < 16; ++k)
                    bfr[k] = *(const v2f*)(bptr + 4 * (half * 16 + k));
                #pragma unroll
                for (int k = 0; k < 16; ++k)
                    acc = __builtin_amdgcn_wmma_f32_16x16x4_f32(
                        /*neg_a=*/false, afrag[half * 16 + k],
                        /*neg_b=*/false, bfr[k],
                        /*c_mod=*/(short)0, acc,
                        /*reuse_a=*/false, /*reuse_b=*/false);
            }
            // D layout: lane L holds col N=L%16, rows M=v (+8 for upper lanes)
            const int cip   = t * 16 + (lane & 15);
            const int mbase = (lane < 16) ? 0 : 8;
            #pragma unroll
            for (int v = 0; v < 8; ++v)
                Ds[mbase + v][cip] = acc[v];
        }
        __syncthreads();

        // ================= scan phase: running top-9 per thread ============
        const int cbase = seg * 64;
        #pragma unroll 1
        for (int u = 0; u < 64; u += 4) {
            const v4f dv = *(const v4f*)&Ds[row][cbase + u];
            const v4f sv = *(const v4f*)&sq[pb + cbase + u];
            #pragma unroll
            for (int e = 0; e < 4; ++e) {
                const int jj = pb + cbase + u + e;
                const float s = sv[e] - 2.0f * dv[e];
                if (jj != grow && s < bv[KNN - 1]) {
                    bv[KNN] = s; bi[KNN] = jj;  // append then one bubble pass
                    #pragma unroll
                    for (int q = KNN; q > 0; --q) {
                        if (bv[q] < bv[q - 1]) {
                            float tv = bv[q]; bv[q] = bv[q - 1]; bv[q - 1] = tv;
                            int   ti = bi[q]; bi[q] = bi[q - 1]; bi[q - 1] = ti;
                        }
                    }
                }
            }
        }
        __syncthreads();   // protect Ds before next panel's compute
    }

    // ================= merge: 8 partial top-9 lists per row ================
    #pragma unroll
    for (int q = 0; q < KNN; ++q) {
        mv[row][seg * KNN + q] = bv[q];
        mi[row][seg * KNN + q] = bi[q];
    }
    __syncthreads();
    if (tid < 16) {
        const int r = tid;
        for (int slot = 0; slot < KNN; ++slot) {
            float best = 3.4e38f; int bidx = 0, bpos = 0;
            for (int q = 0; q < 8 * KNN; ++q) {
                const float vv = mv[r][q];
                if (vv < best) { best = vv; bidx = mi[r][q]; bpos = q; }
            }
            mv[r][bpos] = 3.4e38f;
            nbr[(rbase + r) * KNN + slot] = bidx;
        }
    }
}

// ---------------------------------------------------------------------------
// Kernel 3: max-relative aggregation + concat into y = [x2 | max_j(x2_j)-x2].
// One block of 128 threads per row. Max over neighbors is order-invariant.
// ---------------------------------------------------------------------------
__global__ void mrconv_agg_kernel(const float* __restrict__ x2,
                                  const int* __restrict__ nbr,
                                  float* __restrict__ y) {
    const int n = blockIdx.x;
    const int c = threadIdx.x;
    const float xv = x2[n * C_IN + c];
    float mx = -3.0e38f;
    #pragma unroll
    for (int k = 0; k < KNN; ++k) {
        const int j = nbr[n * KNN + k];
        mx = fmaxf(mx, x2[j * C_IN + c]);
    }
    y[n * C_OUT + c] = xv;
    y[n * C_OUT + C_IN + c] = mx - xv;
}

// ---------------------------------------------------------------------------
// Kernel 4: out = y[16384,256] @ W[256,256] + b via fp32 WMMA.
// One wave per 16x16 output tile; 64 k-steps in 4 batched groups of 16 so the
// A/W fragment loads pipeline ahead of the WMMA chain.
// ---------------------------------------------------------------------------
__launch_bounds__(32)
__global__ void mrconv_gemm_kernel(const float* __restrict__ y,
                                   const float* __restrict__ W,
                                   const float* __restrict__ bvec,
                                   float* __restrict__ out) {
    const int rb   = blockIdx.x * 16;
    const int cb   = blockIdx.y * 16;
    const int lane = threadIdx.x;
    const int col  = cb + (lane & 15);

    const float* aptr = y + (rb + (lane & 15)) * C_OUT + ((lane < 16) ? 0 : 2);
    const int koff = (lane < 16) ? 0 : 2;

    v8f acc = {};
    #pragma unroll
    for (int g = 0; g < 4; ++g) {
        v2f a[16];
        v2f bf[16];
        #pragma unroll
        for (int k = 0; k < 16; ++k) {
            const int ks = g * 16 + k;
            a[k] = *(const v2f*)(aptr + 4 * ks);
            const int kk = 4 * ks + koff;
            bf[k].x = W[kk * C_OUT + col];
            bf[k].y = W[(kk + 1) * C_OUT + col];
        }
        #pragma unroll
        for (int k = 0; k < 16; ++k)
            acc = __builtin_amdgcn_wmma_f32_16x16x4_f32(
                false, a[k], false, bf[k], (short)0, acc, false, false);
    }
    const float bias = bvec[col];
    const int mbase = rb + ((lane < 16) ? 0 : 8);
    #pragma unroll
    for (int v = 0; v < 8; ++v)
        out[(mbase + v) * C_OUT + col] = acc[v] + bias;
}

// ---------------------------------------------------------------------------
extern "C" void kernel_launch(void* const* d_in, const int* in_sizes, int n_in,
                              void* d_out, int out_size, void* d_ws, size_t ws_size,
                              hipStream_t stream) {
    const float* x   = (const float*)d_in[0];
    const float* rpt = (const float*)d_in[1];
    const float* W   = (const float*)d_in[2];
    const float* b   = (const float*)d_in[3];
    float* out = (float*)d_out;

    float* x2  = (float*)d_ws;                                // N*C floats
    float* sq  = x2 + (size_t)N_PTS * C_IN;                   // N floats
    int*   nbr = (int*)(sq + N_PTS);                          // N*K ints
    float* y   = (float*)(nbr + (size_t)N_PTS * KNN);         // N*2C floats

    mrconv_prep_kernel<<<N_PTS, C_IN, 0, stream>>>(x, rpt, x2, sq);
    mrconv_knn_kernel<<<N_PTS / 16, 128, 0, stream>>>(x, sq, nbr);
    mrconv_agg_kernel<<<N_PTS, C_IN, 0, stream>>>(x2, nbr, y);
    mrconv_gemm_kernel<<<dim3(N_PTS / 16, C_OUT / 16), 32, 0, stream>>>(y, W, b, out);
}